// Local_Mask_75746043232890
// MI455X (gfx1250) — compile-verified
//
#include <hip/hip_runtime.h>
#include <stdint.h>

// Local_Mask drop-block kernel for gfx1250 (MI455X).
// Bandwidth-bound: ~411 MB total traffic -> ~18us floor @ 23.3 TB/s.
// gfx1250-specific paths: global_load_async_to_lds_b128 (ASYNCcnt) staging,
// s_wait_asynccnt drain, wave32 shuffle argmax reduction, NT streaming stores.

#define HDIM   56
#define HW     3136          // 56*56
#define NCHUNK 784           // HW/4 float4 chunks per channel
#define TPB    256
#define NWAVES (TPB / 32)
#define HALF   3             // DROP_BLOCK // 2
#define TAIL   (NCHUNK - 3 * TPB)   // 16 chunks in the k=3 tail

typedef float f4 __attribute__((ext_vector_type(4)));  // native vector: NT-store OK

__global__ __launch_bounds__(TPB) void local_mask_kernel(
    const float* __restrict__ x, const int* __restrict__ T,
    float* __restrict__ out)
{
  const int    ch   = blockIdx.x;           // 0 .. B*C-1
  const int    t    = threadIdx.x;          // 0 .. 255
  const size_t base = (size_t)ch * HW;      // element offset of this channel

  __shared__ __align__(16) float tile[HW];  // 12544 B staging buffer
  __shared__ float s_m[NWAVES];
  __shared__ int   s_i[NWAVES];
  __shared__ int   s_box[4];
  __shared__ float s_lam;
  __shared__ int   s_marked;

  // ---- Stage the whole channel into LDS with async 128-bit copies ----
  {
    const uint64_t gbase = (uint64_t)(uintptr_t)(x + base);
    const uint32_t lbase = (uint32_t)(uintptr_t)(&tile[0]); // low 32 bits = LDS byte addr
#pragma unroll
    for (int k = 0; k < 3; ++k) {           // k=0..2 always in-bounds (767 < 784)
      uint32_t voff  = (uint32_t)(t + k * TPB) * 16u;
      uint32_t laddr = lbase + voff;
      asm volatile("global_load_async_to_lds_b128 %0, %1, %2"
                   :: "v"(laddr), "v"(voff), "s"(gbase) : "memory");
    }
    if (t < TAIL) {                         // k=3 tail: chunks 768..783
      uint32_t voff  = (uint32_t)(t + 3 * TPB) * 16u;
      uint32_t laddr = lbase + voff;
      asm volatile("global_load_async_to_lds_b128 %0, %1, %2"
                   :: "v"(laddr), "v"(voff), "s"(gbase) : "memory");
    }
    asm volatile("s_wait_asynccnt 0" ::: "memory");  // drain this wave's async copies
  }
  __syncthreads();                                   // publish LDS across all 8 waves

  // ---- Pass 1: per-thread argmax (first occurrence) over register-resident data ----
  f4    vals[4];
  float m  = -__builtin_inff();
  int   mi = 0x7fffffff;
#pragma unroll
  for (int k = 0; k < 3; ++k) {
    int j = t + k * TPB;
    f4 v = *(const f4*)&tile[4 * j];
    vals[k] = v;
    int e0 = 4 * j;
    if (v.x > m) { m = v.x; mi = e0;     }
    if (v.y > m) { m = v.y; mi = e0 + 1; }
    if (v.z > m) { m = v.z; mi = e0 + 2; }
    if (v.w > m) { m = v.w; mi = e0 + 3; }
  }
  if (t < TAIL) {
    int j = t + 3 * TPB;
    f4 v = *(const f4*)&tile[4 * j];
    vals[3] = v;
    int e0 = 4 * j;
    if (v.x > m) { m = v.x; mi = e0;     }
    if (v.y > m) { m = v.y; mi = e0 + 1; }
    if (v.z > m) { m = v.z; mi = e0 + 2; }
    if (v.w > m) { m = v.w; mi = e0 + 3; }
  }

  // wave32 reduction: max value, ties -> smallest flat index (jnp.argmax semantics)
#pragma unroll
  for (int off = 16; off > 0; off >>= 1) {
    float om = __shfl_xor(m,  off, 32);
    int   oi = __shfl_xor(mi, off, 32);
    if (om > m || (om == m && oi < mi)) { m = om; mi = oi; }
  }
  if ((t & 31) == 0) { s_m[t >> 5] = m; s_i[t >> 5] = mi; }
  __syncthreads();

  if (t == 0) {
    float bm = s_m[0];
    int   bi = s_i[0];
#pragma unroll
    for (int w = 1; w < NWAVES; ++w) {
      float om = s_m[w]; int oi = s_i[w];
      if (om > bm || (om == bm && oi < bi)) { bm = om; bi = oi; }
    }
    // reference uses H for both div and mod (H == W == 56 here)
    int r = bi / HDIM;
    int c = bi % HDIM;
    int h1 = (r - HALF > 0) ? (r - HALF) : 0;
    int h2 = (r + HALF < HDIM - 1) ? (r + HALF) : (HDIM - 1);
    int w1 = (c - HALF > 0) ? (c - HALF) : 0;
    int w2 = (c + HALF < HDIM - 1) ? (c + HALF) : (HDIM - 1);
    s_box[0] = h1; s_box[1] = h2; s_box[2] = w1; s_box[3] = w2;
    int area = (h2 - h1) * (w2 - w1);           // #zeros in S (exclusive box)
    s_lam    = (float)HW / (float)(HW - area);  // HW - area exact in int
    s_marked = (T[ch] > 0) ? 1 : 0;
  }
  __syncthreads();

  const int   h1 = s_box[0], h2 = s_box[1], w1 = s_box[2], w2 = s_box[3];
  const float lam    = s_lam;
  const int   marked = s_marked;

  // ---- Pass 2: scale and store (single HBM write, NT streaming 16B stores) ----
#pragma unroll
  for (int k = 0; k < 4; ++k) {
    int j = t + k * TPB;
    bool valid = (k < 3) || (t < TAIL);
    if (valid) {
      f4    v    = vals[k];
      float o[4] = { v.x, v.y, v.z, v.w };
      int   e0   = 4 * j;
#pragma unroll
      for (int e = 0; e < 4; ++e) {
        int  idx   = e0 + e;
        int  r     = idx / HDIM;
        int  c     = idx % HDIM;
        bool inbox = (r >= h1) && (r < h2) && (c >= w1) && (c < w2);
        float scale = marked ? (inbox ? 0.0f : lam) : 1.0f;
        o[e] *= scale;
      }
      f4 ov = { o[0], o[1], o[2], o[3] };
      __builtin_nontemporal_store(ov, (f4*)&out[base + e0]);
    }
  }
}

extern "C" void kernel_launch(void* const* d_in, const int* in_sizes, int n_in,
                              void* d_out, int out_size, void* d_ws, size_t ws_size,
                              hipStream_t stream) {
  const float* x = (const float*)d_in[0];   // (32, 512, 56, 56) f32
  const int*   T = (const int*)d_in[1];     // (32, 512, 1, 1)  i32
  float*       o = (float*)d_out;           // same shape as x
  const int nch  = in_sizes[1];             // 32*512 = 16384 channels
  local_mask_kernel<<<nch, TPB, 0, stream>>>(x, T, o);
}